// SquaredEuclidean_20658792694342
// MI455X (gfx1250) — compile-verified
//
#include <hip/hip_runtime.h>

typedef __attribute__((ext_vector_type(2))) float v2f;
typedef __attribute__((ext_vector_type(4))) float v4f;
typedef __attribute__((ext_vector_type(8))) float v8f;

#define NROWS 8192
#define KDIM  64

// ---------------------------------------------------------------------------
// Pre-kernel: squared L2 norm of every row of x1 and x2 -> ws[0..16383]
// ws[i]        = ||x1_i||^2   (i < 8192)
// ws[8192 + i] = ||x2_i||^2
// ---------------------------------------------------------------------------
__global__ __launch_bounds__(256) void row_norms_kernel(
    const float* __restrict__ x1, const float* __restrict__ x2,
    float* __restrict__ sq) {
  const int i = blockIdx.x * 256 + threadIdx.x;      // 0..16383
  const float* src = (i < NROWS) ? x1 : x2;
  const int row = (i < NROWS) ? i : (i - NROWS);
  const v4f* p = (const v4f*)(src + (size_t)row * KDIM);
  float s = 0.0f;
#pragma unroll
  for (int j = 0; j < KDIM / 4; ++j) {
    v4f v = p[j];
    s = fmaf(v.x, v.x, s);
    s = fmaf(v.y, v.y, s);
    s = fmaf(v.z, v.z, s);
    s = fmaf(v.w, v.w, s);
  }
  sq[i] = s;
}

// ---------------------------------------------------------------------------
// Main kernel: d2 = max(sq1[m] + sq2[n] - 2 * (x1 . x2^T), 0)
// Workgroup (8 waves) computes a 128x128 output tile.
// Wave w owns M rows [m_blk, m_blk+16); sweeps 8 N-subtiles of 16.
// cross-term via v_wmma_f32_16x16x4_f32 chained over K=64 (16 steps).
//
// A 16x4 f32 fragment (ISA 7.12.2): lane l -> row M = l%16;
//   VGPR v at half h=l/16 holds K = 2h+v  => contiguous float2 @ 4k+2h.
// B 4x16 f32 fragment mirrors it: B[k,n] = x2[n,k] => same float2 pattern
//   from x2 row (n0 + l%16).
// C/D: acc[r] @ lane l -> (M = r + 8h, N = l%16).
// ---------------------------------------------------------------------------
__global__ __launch_bounds__(256) void cdist2_wmma_kernel(
    const float* __restrict__ x1, const float* __restrict__ x2,
    const float* __restrict__ sq, float* __restrict__ out) {
  const int lane = threadIdx.x & 31;
  const int wave = threadIdx.x >> 5;   // 0..7
  const int half = lane >> 4;          // 0 or 1
  const int lidx = lane & 15;          // 0..15

  const int m_blk = blockIdx.y * 128 + wave * 16;  // this wave's M tile
  const int n_blk = blockIdx.x * 128;              // workgroup N tile

  const float* __restrict__ sq1 = sq;
  const float* __restrict__ sq2 = sq + NROWS;

  // Load the A fragments once (16 K-chunks x float2 = 32 VGPRs), reuse 8x.
  const float* arow = x1 + (size_t)(m_blk + lidx) * KDIM + 2 * half;
  v2f a[16];
#pragma unroll
  for (int k = 0; k < 16; ++k)
    a[k] = *(const v2f*)(arow + 4 * k);

#pragma unroll 1
  for (int j = 0; j < 8; ++j) {
    const int n0 = n_blk + j * 16;
    const float* brow = x2 + (size_t)(n0 + lidx) * KDIM + 2 * half;

    // Stage all B fragments, then issue the WMMA chain back-to-back.
    v2f b[16];
#pragma unroll
    for (int k = 0; k < 16; ++k)
      b[k] = *(const v2f*)(brow + 4 * k);

    v8f acc = {};
#pragma unroll
    for (int k = 0; k < 16; ++k) {
      // 8 args: (neg_a, A, neg_b, B, c_mod, C, reuse_a, reuse_b)
      acc = __builtin_amdgcn_wmma_f32_16x16x4_f32(
          false, a[k], false, b[k], (short)0, acc, false, false);
    }

    // Epilogue: d2 = sq1[m] + sq2[n] - 2*cross, clamped at 0.
    const float s2 = sq2[n0 + lidx];
#pragma unroll
    for (int r = 0; r < 8; ++r) {
      const int row = m_blk + r + 8 * half;
      float d2 = fmaf(-2.0f, acc[r], sq1[row] + s2);
      d2 = fmaxf(d2, 0.0f);
      out[(size_t)row * NROWS + (n0 + lidx)] = d2;
    }
  }
}

// ---------------------------------------------------------------------------
extern "C" void kernel_launch(void* const* d_in, const int* in_sizes, int n_in,
                              void* d_out, int out_size, void* d_ws, size_t ws_size,
                              hipStream_t stream) {
  const float* x1 = (const float*)d_in[0];
  const float* x2 = (const float*)d_in[1];
  float* out = (float*)d_out;
  float* sq = (float*)d_ws;   // 2*8192 floats = 64 KB of workspace

  // 1) row norms: 16384 rows total, 256 threads/block
  row_norms_kernel<<<dim3((2 * NROWS) / 256), dim3(256), 0, stream>>>(x1, x2, sq);

  // 2) WMMA GEMM + epilogue: 64x64 workgroups of 128x128 tiles
  cdist2_wmma_kernel<<<dim3(NROWS / 128, NROWS / 128), dim3(256), 0, stream>>>(
      x1, x2, sq, out);
}